// GCNet_3710851744039
// MI455X (gfx1250) — compile-verified
//
#include <hip/hip_runtime.h>
#include <math.h>

typedef __attribute__((ext_vector_type(2))) float v2f;
typedef __attribute__((ext_vector_type(8))) float v8f;

#define N_NODES   10000
#define N_EDGES   160000
#define DIM       512
#define N_LAYERS  3
#define N_CLASSES 10
#define N_GRAPHS  64

// ---------------------------------------------------------------------------
// H = X (10000x512) @ W (512x512), fp32 WMMA 16x16x4.
// One wave -> 5 M-tiles (80 rows) x one 16-col N-tile; one B fragment is
// reused by 5 WMMAs. 10000 = 125 * 80, so 125 M-blocks x 32 N-tiles
// = 4000 waves = 500 blocks x 8 waves, EXEC all-ones everywhere.
// ---------------------------------------------------------------------------
__global__ __launch_bounds__(256) void gemm_wmma_f32(const float* __restrict__ X,
                                                     const float* __restrict__ W,
                                                     float* __restrict__ H) {
    const int lane = threadIdx.x & 31;
    const int wave = threadIdx.x >> 5;
    const int wid  = blockIdx.x * 8 + wave;   // 0..3999
    const int mb   = wid >> 5;                // 125 M-blocks (80 rows each)
    const int n0   = (wid & 31) << 4;         // 32 N-tiles
    const int hi   = lane >> 4;               // K half: 0 -> K{0,1}, 1 -> K{2,3}
    const int lo   = lane & 15;
    const int m0   = mb * 80;

    // A: row m0+lo (+ tile offsets of 16 rows); K pair selected by hi
    const float* xrow = X + (size_t)(m0 + lo) * DIM + 2 * hi;
    // B: column n0+lo; rows k+2*hi, k+2*hi+1
    const float* wcol = W + (size_t)(2 * hi) * DIM + n0 + lo;

    v8f c0 = {}, c1 = {}, c2 = {}, c3 = {}, c4 = {};
#pragma unroll 4
    for (int k = 0; k < DIM; k += 4) {
        v2f b;
        b.x = wcol[(size_t)k * DIM];
        b.y = wcol[(size_t)k * DIM + DIM];
        v2f a0 = *(const v2f*)(xrow + k);
        v2f a1 = *(const v2f*)(xrow + k + 16 * DIM);
        v2f a2 = *(const v2f*)(xrow + k + 32 * DIM);
        v2f a3 = *(const v2f*)(xrow + k + 48 * DIM);
        v2f a4 = *(const v2f*)(xrow + k + 64 * DIM);
        c0 = __builtin_amdgcn_wmma_f32_16x16x4_f32(false, a0, false, b, (short)0, c0, false, false);
        c1 = __builtin_amdgcn_wmma_f32_16x16x4_f32(false, a1, false, b, (short)0, c1, false, false);
        c2 = __builtin_amdgcn_wmma_f32_16x16x4_f32(false, a2, false, b, (short)0, c2, false, false);
        c3 = __builtin_amdgcn_wmma_f32_16x16x4_f32(false, a3, false, b, (short)0, c3, false, false);
        c4 = __builtin_amdgcn_wmma_f32_16x16x4_f32(false, a4, false, b, (short)0, c4, false, false);
    }

    // C/D layout: VGPR r -> M = r + 8*hi, N = n0 + lo
    float* hout = H + (size_t)(m0 + 8 * hi) * DIM + n0 + lo;
#pragma unroll
    for (int r = 0; r < 8; ++r) hout[(size_t)(r) * DIM] = c0[r];
#pragma unroll
    for (int r = 0; r < 8; ++r) hout[(size_t)(r + 16) * DIM] = c1[r];
#pragma unroll
    for (int r = 0; r < 8; ++r) hout[(size_t)(r + 32) * DIM] = c2[r];
#pragma unroll
    for (int r = 0; r < 8; ++r) hout[(size_t)(r + 48) * DIM] = c3[r];
#pragma unroll
    for (int r = 0; r < 8; ++r) hout[(size_t)(r + 64) * DIM] = c4[r];
}

// ---------------------------------------------------------------------------
// Degree with self-loops: deg[col]++ for every edge, +1 per node.
// ---------------------------------------------------------------------------
__global__ void deg_kernel(const int* __restrict__ eidx, float* __restrict__ deg) {
    int e = blockIdx.x * blockDim.x + threadIdx.x;
    const int total = N_EDGES + N_NODES;
    if (e >= total) return;
    int c = (e < N_EDGES) ? eidx[N_EDGES + e] : (e - N_EDGES);
    atomicAdd(&deg[c], 1.0f);
}

__global__ void dinv_kernel(const float* __restrict__ deg, float* __restrict__ dinv) {
    int i = blockIdx.x * blockDim.x + threadIdx.x;
    if (i < N_NODES) dinv[i] = rsqrtf(deg[i]);
}

// ---------------------------------------------------------------------------
// out[col] += h[row] * dinv[row]*dinv[col]  (128 threads per edge, float4)
// ---------------------------------------------------------------------------
__global__ __launch_bounds__(256) void scatter_kernel(const float* __restrict__ h,
                                                      const int* __restrict__ eidx,
                                                      const float* __restrict__ dinv,
                                                      float* __restrict__ out) {
    int e  = blockIdx.x * 2 + (threadIdx.x >> 7);
    int c4 = (threadIdx.x & 127) << 2;
    if (e >= N_EDGES + N_NODES) return;
    int r, c;
    if (e < N_EDGES) { r = eidx[e]; c = eidx[N_EDGES + e]; }
    else             { r = c = e - N_EDGES; }
    float nrm = dinv[r] * dinv[c];
    const float4 hv = *(const float4*)(h + (size_t)r * DIM + c4);
    float* o = out + (size_t)c * DIM + c4;
    atomicAdd(o + 0, hv.x * nrm);
    atomicAdd(o + 1, hv.y * nrm);
    atomicAdd(o + 2, hv.z * nrm);
    atomicAdd(o + 3, hv.w * nrm);
}

__global__ void bias_relu_kernel(const float* __restrict__ agg,
                                 const float* __restrict__ b,
                                 float* __restrict__ x) {
    int i = blockIdx.x * blockDim.x + threadIdx.x;
    if (i >= N_NODES * DIM) return;
    float v = agg[i] + b[i & (DIM - 1)];
    x[i] = v > 0.0f ? v : 0.0f;
}

// ---------------------------------------------------------------------------
// pooled[batch[n]] += x[n]   (128 threads per node, float4)
// ---------------------------------------------------------------------------
__global__ void pool_kernel(const float* __restrict__ x, const int* __restrict__ batch,
                            float* __restrict__ pooled) {
    int t = blockIdx.x * blockDim.x + threadIdx.x;
    if (t >= N_NODES * 128) return;
    int node = t >> 7;
    int c4   = (t & 127) << 2;
    int g = batch[node];
    const float4 v = *(const float4*)(x + (size_t)node * DIM + c4);
    float* p = pooled + (size_t)g * DIM + c4;
    atomicAdd(p + 0, v.x);
    atomicAdd(p + 1, v.y);
    atomicAdd(p + 2, v.z);
    atomicAdd(p + 3, v.w);
}

__global__ void linear_kernel(const float* __restrict__ pooled,
                              const float* __restrict__ lw,
                              const float* __restrict__ lb,
                              float* __restrict__ logits) {
    int t = blockIdx.x * blockDim.x + threadIdx.x;
    if (t >= N_GRAPHS * N_CLASSES) return;
    int g = t / N_CLASSES, c = t % N_CLASSES;
    float acc = lb[c];
    for (int k = 0; k < DIM; ++k)
        acc += pooled[(size_t)g * DIM + k] * lw[(size_t)k * N_CLASSES + c];
    logits[t] = acc;
}

__global__ void logsoftmax_kernel(const float* __restrict__ logits,
                                  float* __restrict__ outp) {
    int g = blockIdx.x * blockDim.x + threadIdx.x;
    if (g >= N_GRAPHS) return;
    float mx = -1e30f;
    for (int c = 0; c < N_CLASSES; ++c) mx = fmaxf(mx, logits[g * N_CLASSES + c]);
    float s = 0.0f;
    for (int c = 0; c < N_CLASSES; ++c) s += expf(logits[g * N_CLASSES + c] - mx);
    float ls = logf(s);
    for (int c = 0; c < N_CLASSES; ++c)
        outp[g * N_CLASSES + c] = logits[g * N_CLASSES + c] - mx - ls;
}

// ---------------------------------------------------------------------------
extern "C" void kernel_launch(void* const* d_in, const int* in_sizes, int n_in,
                              void* d_out, int out_size, void* d_ws, size_t ws_size,
                              hipStream_t stream) {
    const float* x_in  = (const float*)d_in[0];
    const int*   eidx  = (const int*)d_in[1];   // [2, E]
    const int*   batch = (const int*)d_in[2];   // [N]
    const float* Ws    = (const float*)d_in[3]; // [3, 512, 512]
    const float* bs    = (const float*)d_in[4]; // [3, 512]
    const float* lw    = (const float*)d_in[5]; // [512, 10]
    const float* lb    = (const float*)d_in[6]; // [10]
    float* outp = (float*)d_out;

    char* ws = (char*)d_ws;
    const size_t featB = (size_t)N_NODES * DIM * sizeof(float); // 20.48 MB
    float* X      = (float*)(ws);
    float* H      = (float*)(ws + featB);
    float* AGG    = (float*)(ws + 2 * featB);
    float* deg    = (float*)(ws + 3 * featB);
    float* dinv   = (float*)(ws + 3 * featB + (64u << 10));
    float* pooled = (float*)(ws + 3 * featB + (128u << 10));
    float* logits = (float*)(ws + 3 * featB + (128u << 10) + (size_t)N_GRAPHS * DIM * sizeof(float));

    // X <- input features (never mutate d_in)
    hipMemcpyAsync(X, x_in, featB, hipMemcpyDeviceToDevice, stream);

    // Normalization: deg^-1/2 with self-loops (graph fixed across layers)
    hipMemsetAsync(deg, 0, N_NODES * sizeof(float), stream);
    deg_kernel<<<(N_EDGES + N_NODES + 255) / 256, 256, 0, stream>>>(eidx, deg);
    dinv_kernel<<<(N_NODES + 255) / 256, 256, 0, stream>>>(deg, dinv);

    for (int layer = 0; layer < N_LAYERS; ++layer) {
        gemm_wmma_f32<<<500, 256, 0, stream>>>(X, Ws + (size_t)layer * DIM * DIM, H);
        hipMemsetAsync(AGG, 0, featB, stream);
        scatter_kernel<<<(N_EDGES + N_NODES) / 2, 256, 0, stream>>>(H, eidx, dinv, AGG);
        bias_relu_kernel<<<(N_NODES * DIM + 255) / 256, 256, 0, stream>>>(AGG, bs + layer * DIM, X);
    }

    hipMemsetAsync(pooled, 0, (size_t)N_GRAPHS * DIM * sizeof(float), stream);
    pool_kernel<<<(N_NODES * 128 + 255) / 256, 256, 0, stream>>>(X, batch, pooled);
    linear_kernel<<<(N_GRAPHS * N_CLASSES + 63) / 64, 64, 0, stream>>>(pooled, lw, lb, logits);
    logsoftmax_kernel<<<1, 64, 0, stream>>>(logits, outp);
}